// PixAdvLoss_20615843020868
// MI455X (gfx1250) — compile-verified
//
#include <hip/hip_runtime.h>

// Problem constants (match reference)
#define B_  4
#define C_  19
#define H_  512
#define W_  1024
#define HW_ (H_ * W_)                       // 524288
#define CHUNKS (HW_ / 4)                    // float4 chunks per sample = 131072
#define THREADS 256
#define BLOCKS_X (CHUNKS / THREADS)         // 512
#define NPART (BLOCKS_X * B_)               // 2048 per-block partials
#define INV_HW (1.0f / (float)HW_)
#define INV_N  (1.0f / (float)(B_ * HW_))

typedef __attribute__((ext_vector_type(2))) float v2f;
typedef __attribute__((ext_vector_type(8))) float v8f;

// ---------------------------------------------------------------- init counts
__global__ void pixadv_init(unsigned* __restrict__ counts) {
    int i = threadIdx.x;
    if (i < B_ * C_) counts[i] = 0u;
}

// ---------------------------------------------------------------- histogram
__global__ __launch_bounds__(THREADS) void pixadv_hist(const int* __restrict__ labels,
                                                       unsigned* __restrict__ counts) {
    __shared__ unsigned hist[8][32];        // per-wave private histograms (padded)
    const int tid  = threadIdx.x;
    const int wave = tid >> 5;
    const int b    = blockIdx.y;

    for (int i = tid; i < 8 * 32; i += THREADS) ((unsigned*)hist)[i] = 0u;
    __syncthreads();

    const int idx = blockIdx.x * THREADS + tid;
    const int4 l  = reinterpret_cast<const int4*>(labels + (size_t)b * HW_)[idx];
    atomicAdd(&hist[wave][l.x], 1u);
    atomicAdd(&hist[wave][l.y], 1u);
    atomicAdd(&hist[wave][l.z], 1u);
    atomicAdd(&hist[wave][l.w], 1u);
    __syncthreads();

    if (tid < C_) {
        unsigned s = 0;
#pragma unroll
        for (int wv = 0; wv < 8; ++wv) s += hist[wv][tid];
        atomicAdd(&counts[b * C_ + tid], s);   // integer atomic: deterministic
    }
}

// ---------------------------------------------------------------- main fused loss
__global__ __launch_bounds__(THREADS) void pixadv_loss(const float* __restrict__ parser,
                                                       const float* __restrict__ disc,
                                                       const int*   __restrict__ labels,
                                                       const unsigned* __restrict__ counts,
                                                       float* __restrict__ partials) {
    __shared__ float bal[32];
    __shared__ float wsum[8];

    const int tid = threadIdx.x;
    const int b   = blockIdx.y;

    // per-sample class balance weights: 1 - count/HW  (HW is a power of two -> exact)
    if (tid < C_) bal[tid] = 1.0f - (float)counts[b * C_ + tid] * INV_HW;
    __syncthreads();

    const int idx = blockIdx.x * THREADS + tid;        // float4 chunk index in sample
    const float4* pp = reinterpret_cast<const float4*>(parser + (size_t)b * C_ * HW_) + idx;
    const int4   lab = reinterpret_cast<const int4*>(labels + (size_t)b * HW_)[idx];
    const float4 dsc = reinterpret_cast<const float4*>(disc + (size_t)b * HW_)[idx];

    const int stride4 = HW_ / 4;                        // float4 stride between channel planes

    float s0 = 0.f, s1 = 0.f, s2 = 0.f, s3 = 0.f;       // sum of exp(x)
    float xl0 = 0.f, xl1 = 0.f, xl2 = 0.f, xl3 = 0.f;   // x at the label channel

    // warm the streams a few planes ahead (global_prefetch_b8)
    __builtin_prefetch(&pp[(size_t)2 * stride4], 0, 0);
    __builtin_prefetch(&pp[(size_t)3 * stride4], 0, 0);

#pragma unroll
    for (int c = 0; c < C_; ++c) {
        const float4 x = pp[(size_t)c * stride4];
        if (c + 4 < C_) __builtin_prefetch(&pp[(size_t)(c + 4) * stride4], 0, 0);
        s0 += __expf(x.x); s1 += __expf(x.y); s2 += __expf(x.z); s3 += __expf(x.w);
        xl0 = (lab.x == c) ? x.x : xl0;
        xl1 = (lab.y == c) ? x.y : xl1;
        xl2 = (lab.z == c) ? x.z : xl2;
        xl3 = (lab.w == c) ? x.w : xl3;
    }

    // domain_shift = logsumexp(x) - x[label]
    const float dom0 = __logf(s0) - xl0;
    const float dom1 = __logf(s1) - xl1;
    const float dom2 = __logf(s2) - xl2;
    const float dom3 = __logf(s3) - xl3;

    // fooling_loss = softplus(d) = max(d,0) + log1p(exp(-|d|))
    const float f0 = fmaxf(dsc.x, 0.f) + log1pf(__expf(-fabsf(dsc.x)));
    const float f1 = fmaxf(dsc.y, 0.f) + log1pf(__expf(-fabsf(dsc.y)));
    const float f2 = fmaxf(dsc.z, 0.f) + log1pf(__expf(-fabsf(dsc.z)));
    const float f3 = fmaxf(dsc.w, 0.f) + log1pf(__expf(-fabsf(dsc.w)));

    float p = f0 * dom0 * bal[lab.x]
            + f1 * dom1 * bal[lab.y]
            + f2 * dom2 * bal[lab.z]
            + f3 * dom3 * bal[lab.w];

    // ---- wave32 reduction via V_WMMA_F32_16X16X4_F32 ----
    // A[16x4] holds each lane's partial exactly once (rest zero); B = ones(4x16)
    // => D[m,n] = row-sum. Sum of this lane's 8 accumulator rows + shfl_xor(16)
    // gives the exact f32 sum of all 32 lanes' partials (layout-robust).
    v2f a;  a[0] = p;   a[1] = 0.f;
    v2f bo; bo[0] = 1.f; bo[1] = 1.f;
    v8f acc = {};
    acc = __builtin_amdgcn_wmma_f32_16x16x4_f32(false, a, false, bo,
                                                (short)0, acc, false, false);
    float wsv = acc[0] + acc[1] + acc[2] + acc[3] + acc[4] + acc[5] + acc[6] + acc[7];
    wsv += __shfl_xor(wsv, 16, 32);

    const int wave = tid >> 5, lane = tid & 31;
    if (lane == 0) wsum[wave] = wsv;
    __syncthreads();
    if (tid == 0) {
        float t = 0.f;
#pragma unroll
        for (int i = 0; i < 8; ++i) t += wsum[i];
        partials[(size_t)b * gridDim.x + blockIdx.x] = t;   // no float atomics: deterministic
    }
}

// ---------------------------------------------------------------- final mean
__global__ __launch_bounds__(256) void pixadv_final(const float* __restrict__ partials,
                                                    float* __restrict__ out) {
    __shared__ float sm[256];
    const int tid = threadIdx.x;
    float s = 0.f;
#pragma unroll
    for (int i = 0; i < NPART / 256; ++i) s += partials[i * 256 + tid];
    sm[tid] = s;
    __syncthreads();
    for (int off = 128; off > 0; off >>= 1) {
        if (tid < off) sm[tid] += sm[tid + off];
        __syncthreads();
    }
    if (tid == 0) out[0] = sm[0] * INV_N;
}

// ---------------------------------------------------------------- launch
extern "C" void kernel_launch(void* const* d_in, const int* in_sizes, int n_in,
                              void* d_out, int out_size, void* d_ws, size_t ws_size,
                              hipStream_t stream) {
    (void)in_sizes; (void)n_in; (void)out_size; (void)ws_size;

    const float* parser = (const float*)d_in[0];   // [B,C,H,W] f32
    const float* disc   = (const float*)d_in[1];   // [B,1,H,W] f32
    const int*   labels = (const int*)d_in[2];     // [B,H,W]   i32

    unsigned* counts   = (unsigned*)d_ws;                        // 76 * 4 B
    float*    partials = (float*)((char*)d_ws + 512);            // 2048 * 4 B

    pixadv_init<<<1, 128, 0, stream>>>(counts);
    pixadv_hist<<<dim3(BLOCKS_X, B_), THREADS, 0, stream>>>(labels, counts);
    pixadv_loss<<<dim3(BLOCKS_X, B_), THREADS, 0, stream>>>(parser, disc, labels,
                                                            counts, partials);
    pixadv_final<<<1, 256, 0, stream>>>(partials, (float*)d_out);
}